// GroupMambaLayer_71390946394230
// MI455X (gfx1250) — compile-verified
//
#include <hip/hip_runtime.h>
#include <hip/hip_bf16.h>
#include <math.h>

// ---------------------------------------------------------------------------
// GroupMamba (VMamba SS2D) fused layer for gfx1250 (MI455X, wave32, WMMA).
//   C=256, G=4, D=64, NST=1, DTR=4, B=16, H=W=64, L=4096.
// GEMMs use V_WMMA_F32_16X16X4_F32 (exact fp32, 16x16 tile, K-step 4).
// Selective scan: 3-phase chunked decomposition (64 chunks x 64 steps) with
// global_prefetch_b8 look-ahead on the latency-critical u stream.
// ---------------------------------------------------------------------------

#define DEV __device__ __forceinline__

typedef __attribute__((ext_vector_type(2))) float f32x2;
typedef __attribute__((ext_vector_type(8))) float f32x8;

constexpr int Cc   = 256;
constexpr int Gg   = 4;
constexpr int Dd   = 64;
constexpr int Bb   = 16;
constexpr int Hh   = 64;
constexpr int Ww   = 64;
constexpr int Ll   = Hh * Ww;          // 4096
constexpr int NCH  = 64;               // scan chunks
constexpr int CHL  = Ll / NCH;         // 64 steps per chunk
constexpr int PFD  = 8;                // scan prefetch distance (steps)
constexpr float EPS = 1e-5f;

DEV float sigmoidf_(float x) { return 1.0f / (1.0f + __expf(-x)); }
DEV float siluf_(float x)    { return x * sigmoidf_(x); }
DEV float softplusf_(float x){ return x > 20.0f ? x : log1pf(__expf(x)); }

DEV float wave_sum(float v) {
  #pragma unroll
  for (int m = 16; m >= 1; m >>= 1) v += __shfl_xor(v, m, 32);
  return v;
}

// scan index s -> spatial index l, per group's scan direction
DEV int scan2spatial(int g, int s) {
  switch (g) {
    case 0:  return s;                                   // row-major fwd
    case 1:  return ((s & 63) << 6) | (s >> 6);          // col-major fwd
    case 2:  return (Ll - 1) - s;                        // row-major rev
    default: { int j = (Ll - 1) - s;                     // col-major rev
               return ((j & 63) << 6) | (j >> 6); }
  }
}

// ---------------------------------------------------------------------------
// WMMA f32 16x16 tile: C += A(16xK) * B(Kx16), K multiple of 4.
// A fragment: lane m=lane&15 holds K pairs (k+2*(lane>>4), +1).
// B fragment mirrors A with N in lane&15. C/D: vgpr r -> M=r+8*(lane>>4).
// ---------------------------------------------------------------------------
DEV f32x8 wmma_tile_f32(const float* __restrict__ A, int lda,
                        const float* __restrict__ B, int ldb, int K) {
  const int lane = threadIdx.x & 31;
  const int mn = lane & 15;
  const int kh = (lane >> 4) << 1;
  f32x8 acc = {0.f, 0.f, 0.f, 0.f, 0.f, 0.f, 0.f, 0.f};
  for (int k = 0; k < K; k += 4) {
    f32x2 a, b;
    a[0] = A[mn * lda + k + kh];
    a[1] = A[mn * lda + k + kh + 1];
    b[0] = B[(k + kh) * ldb + mn];
    b[1] = B[(k + kh + 1) * ldb + mn];
    acc = __builtin_amdgcn_wmma_f32_16x16x4_f32(false, a, false, b,
                                                (short)0, acc, false, false);
  }
  return acc;
}

// ---------------------------------------------------------------------------
// K1: token LayerNorm over C=256.  1 wave / token, 8 channels / lane.
// ---------------------------------------------------------------------------
__global__ void k_layernorm_c(const float* __restrict__ x,
                              const float* __restrict__ w,
                              const float* __restrict__ b,
                              float* __restrict__ xn) {
  const int warp = threadIdx.x >> 5, lane = threadIdx.x & 31;
  const int tok = blockIdx.x * 8 + warp;               // [0, B*L)
  const float* xr = x + (size_t)tok * Cc;
  float v[8], s = 0.f;
  #pragma unroll
  for (int j = 0; j < 8; ++j) { v[j] = xr[lane + 32 * j]; s += v[j]; }
  const float mean = wave_sum(s) * (1.0f / Cc);
  float q = 0.f;
  #pragma unroll
  for (int j = 0; j < 8; ++j) { float d = v[j] - mean; q += d * d; }
  const float rs = rsqrtf(wave_sum(q) * (1.0f / Cc) + EPS);
  float* o = xn + (size_t)tok * Cc;
  #pragma unroll
  for (int j = 0; j < 8; ++j) {
    int c = lane + 32 * j;
    o[c] = (v[j] - mean) * rs * w[c] + b[c];
  }
}

// ---------------------------------------------------------------------------
// K2a: SE squeeze zp[b,c] = mean_l xn[b,l,c].  1 thread / (b,c).
// Strided (1KB) column walk -> prefetch ahead to keep HBM pipe full.
// ---------------------------------------------------------------------------
__global__ void k_colmean(const float* __restrict__ xn, float* __restrict__ zp) {
  const int idx = blockIdx.x * blockDim.x + threadIdx.x;   // b*C + c
  const int bidx = idx >> 8, c = idx & 255;
  const float* p = xn + (size_t)bidx * Ll * Cc + c;
  float s = 0.f;
  for (int l = 0; l < Ll; ++l) {
    if (l + 32 < Ll) __builtin_prefetch(p + (size_t)(l + 32) * Cc, 0, 0);
    s += p[(size_t)l * Cc];
  }
  zp[idx] = s * (1.0f / Ll);
}

// ---------------------------------------------------------------------------
// K2b: gate = sigmoid(relu(zp@fc1 + b1)@fc2 + b2).  1 block / batch.
// ---------------------------------------------------------------------------
__global__ void k_gate(const float* __restrict__ zp,
                       const float* __restrict__ fc1w, const float* __restrict__ fc1b,
                       const float* __restrict__ fc2w, const float* __restrict__ fc2b,
                       float* __restrict__ gate) {
  __shared__ float s_zp[Cc];
  __shared__ float s_h[16];
  const int bidx = blockIdx.x, t = threadIdx.x;
  s_zp[t] = zp[bidx * Cc + t];
  __syncthreads();
  if (t < 16) {
    float a = fc1b[t];
    for (int c = 0; c < Cc; ++c) a += s_zp[c] * fc1w[c * 16 + t];
    s_h[t] = fmaxf(a, 0.0f);
  }
  __syncthreads();
  float a = fc2b[t];
  #pragma unroll
  for (int j = 0; j < 16; ++j) a += s_h[j] * fc2w[j * Cc + t];
  gate[bidx * Cc + t] = sigmoidf_(a);
}

// ---------------------------------------------------------------------------
// K3: in-proj GEMM per (g,b): xn_g (L x 64) @ in_w_g (64 x 128).
// cols [0,64) -> xc (conv input, group-major layout), cols [64,128) -> silu -> z.
// grid (M/128, 8, G*B), block 256 (8 waves, tiles stacked along M).
// ---------------------------------------------------------------------------
__global__ void k_gemm_inproj(const float* __restrict__ xn,
                              const float* __restrict__ in_w,
                              float* __restrict__ xc, float* __restrict__ z) {
  const int wave = threadIdx.x >> 5, lane = threadIdx.x & 31;
  const int tileM = blockIdx.x * 128 + wave * 16;
  const int tileN = blockIdx.y * 16;
  const int gb = blockIdx.z, g = gb >> 4, bidx = gb & 15;
  const float* A = xn + ((size_t)bidx * Ll + tileM) * Cc + g * Dd;
  const float* Bm = in_w + (size_t)g * Dd * 128 + tileN;
  f32x8 acc = wmma_tile_f32(A, Cc, Bm, 128, Dd);
  const int col = tileN + (lane & 15);
  const int rb = (lane >> 4) * 8;
  const size_t base = (size_t)gb * Ll;
  #pragma unroll
  for (int r = 0; r < 8; ++r) {
    const int l = tileM + rb + r;
    const float v = acc[r];
    if (col < Dd) xc[(base + l) * Dd + col] = v;
    else          z [(base + l) * Dd + (col - Dd)] = siluf_(v);
  }
}

// ---------------------------------------------------------------------------
// K4: depthwise 3x3 SAME conv + bias + SiLU -> u.  1 thread / (g,b,h,w,d).
// ---------------------------------------------------------------------------
__global__ void k_conv_dw(const float* __restrict__ xc,
                          const float* __restrict__ cw,
                          const float* __restrict__ cb,
                          float* __restrict__ u) {
  const int idx = blockIdx.x * blockDim.x + threadIdx.x;
  const int d = idx & 63, wq = (idx >> 6) & 63, hq = (idx >> 12) & 63;
  const int gb = idx >> 18, g = gb >> 4;
  const float* base = xc + (size_t)gb * Ll * Dd;
  float acc = cb[g * Dd + d];
  #pragma unroll
  for (int kh = 0; kh < 3; ++kh) {
    const int hh = hq + kh - 1;
    if (hh < 0 || hh >= Hh) continue;
    #pragma unroll
    for (int kw = 0; kw < 3; ++kw) {
      const int ww2 = wq + kw - 1;
      if (ww2 < 0 || ww2 >= Ww) continue;
      acc += base[((size_t)(hh * Ww + ww2)) * Dd + d] *
             cw[((g * 3 + kh) * 3 + kw) * Dd + d];
    }
  }
  u[(size_t)idx] = siluf_(acc);
}

// ---------------------------------------------------------------------------
// K5: x_dbl = u @ xproj_w^T -> 6 values per (g,b,l).  1 wave / token.
// ---------------------------------------------------------------------------
__global__ void k_xdbl(const float* __restrict__ u,
                       const float* __restrict__ xw,
                       float* __restrict__ dbc) {
  const int warp = threadIdx.x >> 5, lane = threadIdx.x & 31;
  const int tok = blockIdx.x * 8 + warp;               // g*B*L + b*L + l
  const int g = tok >> 16;
  const float* ur = u + (size_t)tok * Dd;
  const float u0 = ur[lane], u1 = ur[lane + 32];
  const float* w = xw + g * 6 * Dd;
  #pragma unroll
  for (int c = 0; c < 6; ++c) {
    float p = u0 * w[c * Dd + lane] + u1 * w[c * Dd + lane + 32];
    p = wave_sum(p);
    if (lane == 0) dbc[(size_t)tok * 6 + c] = p;
  }
}

// ---------------------------------------------------------------------------
// Scan phases. h_t = a_t h_{t-1} + b_t with
//   a = exp(dt * (-exp(A_log[d]))),  b = dt * u * Bs,  dt = softplus(dtr.W + b).
// Phase A: per chunk, compute (prod a, local scan) with h0=0.
// Phase B: sequential over 64 chunks per (g,b,d) -> h0 per chunk.
// Phase C: replay with correct h0; y = h*Cs + u*Dskip, scattered to spatial l.
// ---------------------------------------------------------------------------
struct ScanCtx { int g, bidx, ch, gbL; };

DEV ScanCtx scan_ctx() {
  ScanCtx c;
  c.ch = blockIdx.x & 63;
  int gb = blockIdx.x >> 6;
  c.g = gb >> 4; c.bidx = gb & 15;
  c.gbL = gb * Ll;
  return c;
}

__global__ void k_scan_a(const float* __restrict__ u,
                         const float* __restrict__ dbc,
                         const float* __restrict__ dtw,
                         const float* __restrict__ dtb,
                         const float* __restrict__ alog,
                         float* __restrict__ chP, float* __restrict__ chS) {
  __shared__ float sdbc[CHL * 6];
  __shared__ int sl[CHL];
  const ScanCtx cx = scan_ctx();
  const int d = threadIdx.x;
  sl[d] = scan2spatial(cx.g, cx.ch * CHL + d);
  __syncthreads();
  for (int i = d; i < CHL * 6; i += CHL)
    sdbc[i] = dbc[((size_t)cx.gbL + sl[i / 6]) * 6 + (i % 6)];
  __syncthreads();
  // warm the u stream while the serial loop starts
  #pragma unroll
  for (int i = 0; i < PFD; ++i)
    __builtin_prefetch(&u[((size_t)cx.gbL + sl[i]) * Dd + d], 0, 0);
  const int gd = cx.g * Dd + d;
  const float w0 = dtw[gd * 4 + 0], w1 = dtw[gd * 4 + 1];
  const float w2 = dtw[gd * 4 + 2], w3 = dtw[gd * 4 + 3];
  const float bia = dtb[gd];
  const float An = -__expf(alog[gd]);
  float P = 1.0f, h = 0.0f;
  for (int i = 0; i < CHL; ++i) {
    if (i + PFD < CHL)
      __builtin_prefetch(&u[((size_t)cx.gbL + sl[i + PFD]) * Dd + d], 0, 0);
    const float* t = sdbc + i * 6;
    const float dt = softplusf_(t[0]*w0 + t[1]*w1 + t[2]*w2 + t[3]*w3 + bia);
    const float uu = u[((size_t)cx.gbL + sl[i]) * Dd + d];
    const float a = __expf(dt * An);
    P *= a;
    h = h * a + dt * uu * t[4];
  }
  const int idx = (((cx.g * Bb + cx.bidx) * Dd) + d) * NCH + cx.ch;
  chP[idx] = P; chS[idx] = h;
}

__global__ void k_scan_b(const float* __restrict__ chP,
                         const float* __restrict__ chS,
                         float* __restrict__ h0) {
  const int idx = blockIdx.x * blockDim.x + threadIdx.x;  // (g*B+b)*D + d
  const float* P = chP + (size_t)idx * NCH;
  const float* S = chS + (size_t)idx * NCH;
  float* H = h0 + (size_t)idx * NCH;
  float h = 0.0f;
  for (int c = 0; c < NCH; ++c) { H[c] = h; h = h * P[c] + S[c]; }
}

__global__ void k_scan_c(const float* __restrict__ u,
                         const float* __restrict__ dbc,
                         const float* __restrict__ dtw,
                         const float* __restrict__ dtb,
                         const float* __restrict__ alog,
                         const float* __restrict__ dskip,
                         const float* __restrict__ h0,
                         float* __restrict__ y) {
  __shared__ float sdbc[CHL * 6];
  __shared__ int sl[CHL];
  const ScanCtx cx = scan_ctx();
  const int d = threadIdx.x;
  sl[d] = scan2spatial(cx.g, cx.ch * CHL + d);
  __syncthreads();
  for (int i = d; i < CHL * 6; i += CHL)
    sdbc[i] = dbc[((size_t)cx.gbL + sl[i / 6]) * 6 + (i % 6)];
  __syncthreads();
  #pragma unroll
  for (int i = 0; i < PFD; ++i)
    __builtin_prefetch(&u[((size_t)cx.gbL + sl[i]) * Dd + d], 0, 0);
  const int gd = cx.g * Dd + d;
  const float w0 = dtw[gd * 4 + 0], w1 = dtw[gd * 4 + 1];
  const float w2 = dtw[gd * 4 + 2], w3 = dtw[gd * 4 + 3];
  const float bia = dtb[gd];
  const float An = -__expf(alog[gd]);
  const float dsk = dskip[gd];
  float h = h0[(((cx.g * Bb + cx.bidx) * Dd) + d) * NCH + cx.ch];
  for (int i = 0; i < CHL; ++i) {
    if (i + PFD < CHL)
      __builtin_prefetch(&u[((size_t)cx.gbL + sl[i + PFD]) * Dd + d], 0, 0);
    const float* t = sdbc + i * 6;
    const float dt = softplusf_(t[0]*w0 + t[1]*w1 + t[2]*w2 + t[3]*w3 + bia);
    const size_t ui = ((size_t)cx.gbL + sl[i]) * Dd + d;
    const float uu = u[ui];
    const float a = __expf(dt * An);
    h = h * a + dt * uu * t[4];
    y[ui] = h * t[5] + uu * dsk;   // scatter to spatial position (CrossMerge)
  }
}

// ---------------------------------------------------------------------------
// K9: per-group LayerNorm over D=64 then * silu(z).  In place on y. 1 wave/token.
// ---------------------------------------------------------------------------
__global__ void k_ln_group(float* __restrict__ y, const float* __restrict__ z,
                           const float* __restrict__ ow, const float* __restrict__ ob) {
  const int warp = threadIdx.x >> 5, lane = threadIdx.x & 31;
  const int tok = blockIdx.x * 8 + warp;               // (g*B+b)*L + l
  const int g = tok >> 16;
  float* yr = y + (size_t)tok * Dd;
  const float v0 = yr[lane], v1 = yr[lane + 32];
  const float mean = wave_sum(v0 + v1) * (1.0f / Dd);
  const float d0 = v0 - mean, d1 = v1 - mean;
  const float rs = rsqrtf(wave_sum(d0 * d0 + d1 * d1) * (1.0f / Dd) + EPS);
  const float* zr = z + (size_t)tok * Dd;
  const int gd = g * Dd;
  yr[lane]      = (d0 * rs * ow[gd + lane]      + ob[gd + lane])      * zr[lane];
  yr[lane + 32] = (d1 * rs * ow[gd + lane + 32] + ob[gd + lane + 32]) * zr[lane + 32];
}

// ---------------------------------------------------------------------------
// K10: out-proj GEMM per g: ynorm (B*L x 64) @ out_w_g (64x64) -> ym[:, g*64:].
// ---------------------------------------------------------------------------
__global__ void k_gemm_outproj(const float* __restrict__ yn,
                               const float* __restrict__ ow,
                               float* __restrict__ ym) {
  const int wave = threadIdx.x >> 5, lane = threadIdx.x & 31;
  const int tileM = blockIdx.x * 128 + wave * 16;
  const int tileN = blockIdx.y * 16;
  const int g = blockIdx.z;
  const float* A = yn + ((size_t)g * Bb * Ll + tileM) * Dd;
  const float* Bm = ow + (size_t)g * Dd * Dd + tileN;
  f32x8 acc = wmma_tile_f32(A, Dd, Bm, Dd, Dd);
  const int col = g * Dd + tileN + (lane & 15);
  const int rb = (lane >> 4) * 8;
  #pragma unroll
  for (int r = 0; r < 8; ++r)
    ym[(size_t)(tileM + rb + r) * Cc + col] = acc[r];
}

// ---------------------------------------------------------------------------
// K11: t = ym * skip * xn * gate[b]; LayerNorm over C -> tn.  1 wave / token.
// ---------------------------------------------------------------------------
__global__ void k_scale_ln(const float* __restrict__ ym,
                           const float* __restrict__ xn,
                           const float* __restrict__ gate,
                           const float* __restrict__ skip,
                           const float* __restrict__ w, const float* __restrict__ b,
                           float* __restrict__ tn) {
  const int warp = threadIdx.x >> 5, lane = threadIdx.x & 31;
  const int tok = blockIdx.x * 8 + warp;               // b*L + l
  const int bidx = tok >> 12;
  const float sk = skip[0];
  const float* yr = ym + (size_t)tok * Cc;
  const float* xr = xn + (size_t)tok * Cc;
  const float* gr = gate + bidx * Cc;
  float v[8], s = 0.f;
  #pragma unroll
  for (int j = 0; j < 8; ++j) {
    const int c = lane + 32 * j;
    v[j] = yr[c] * sk * xr[c] * gr[c];
    s += v[j];
  }
  const float mean = wave_sum(s) * (1.0f / Cc);
  float q = 0.f;
  #pragma unroll
  for (int j = 0; j < 8; ++j) { const float d = v[j] - mean; q += d * d; }
  const float rs = rsqrtf(wave_sum(q) * (1.0f / Cc) + EPS);
  float* o = tn + (size_t)tok * Cc;
  #pragma unroll
  for (int j = 0; j < 8; ++j) {
    const int c = lane + 32 * j;
    o[c] = (v[j] - mean) * rs * w[c] + b[c];
  }
}

// ---------------------------------------------------------------------------
// K12: final GEMM (B*L x 256) @ proj_w (256x256) + proj_b -> out.
// ---------------------------------------------------------------------------
__global__ void k_gemm_final(const float* __restrict__ tn,
                             const float* __restrict__ pw,
                             const float* __restrict__ pb,
                             float* __restrict__ out) {
  const int wave = threadIdx.x >> 5, lane = threadIdx.x & 31;
  const int tileM = blockIdx.x * 128 + wave * 16;
  const int tileN = blockIdx.y * 16;
  const float* A = tn + (size_t)tileM * Cc;
  const float* Bm = pw + tileN;
  f32x8 acc = wmma_tile_f32(A, Cc, Bm, Cc, Cc);
  const int col = tileN + (lane & 15);
  const int rb = (lane >> 4) * 8;
  const float bias = pb[col];
  #pragma unroll
  for (int r = 0; r < 8; ++r)
    out[(size_t)(tileM + rb + r) * Cc + col] = acc[r] + bias;
}

// ---------------------------------------------------------------------------
extern "C" void kernel_launch(void* const* d_in, const int* in_sizes, int n_in,
                              void* d_out, int out_size, void* d_ws, size_t ws_size,
                              hipStream_t stream) {
  const float* x       = (const float*)d_in[0];
  const float* norm_w  = (const float*)d_in[3];
  const float* norm_b  = (const float*)d_in[4];
  const float* fc1_w   = (const float*)d_in[5];
  const float* fc1_b   = (const float*)d_in[6];
  const float* fc2_w   = (const float*)d_in[7];
  const float* fc2_b   = (const float*)d_in[8];
  const float* in_w    = (const float*)d_in[9];
  const float* conv_w  = (const float*)d_in[10];
  const float* conv_b  = (const float*)d_in[11];
  const float* xproj_w = (const float*)d_in[12];
  const float* dtproj_w= (const float*)d_in[13];
  const float* dtproj_b= (const float*)d_in[14];
  const float* A_log   = (const float*)d_in[15];
  const float* Dskip   = (const float*)d_in[16];
  const float* onorm_w = (const float*)d_in[17];
  const float* onorm_b = (const float*)d_in[18];
  const float* out_w   = (const float*)d_in[19];
  const float* skip_s  = (const float*)d_in[20];
  const float* proj_w  = (const float*)d_in[21];
  const float* proj_b  = (const float*)d_in[22];
  float* out = (float*)d_out;

  // Workspace layout (floats); big buffers reused: xc->ym, z->tn.
  const size_t SZ = (size_t)Bb * Ll * Cc;            // 16.78M floats
  float* ws   = (float*)d_ws;
  float* xn   = ws;                                   // [K1 .. K11]
  float* xc   = xn + SZ;                              // conv in; reused as ym
  float* zb   = xc + SZ;                              // silu(z); reused as tn
  float* ub   = zb + SZ;                              // conv out (scan input)
  float* yb   = ub + SZ;                              // scan out / ynorm
  float* dbc  = yb + SZ;                              // (g,b,l,6)
  float* chP  = dbc + (size_t)Gg * Bb * Ll * 6;       // 262144
  float* chS  = chP + (size_t)Gg * Bb * Dd * NCH;
  float* h0b  = chS + (size_t)Gg * Bb * Dd * NCH;
  float* zp   = h0b + (size_t)Gg * Bb * Dd * NCH;
  float* gate = zp + Bb * Cc;
  float* ym   = xc;
  float* tn   = zb;

  const int TOK = Bb * Ll;                            // 65536 tokens

  // 1) token LayerNorm
  k_layernorm_c<<<TOK / 8, 256, 0, stream>>>(x, norm_w, norm_b, xn);
  // 2) SE gate
  k_colmean<<<(Bb * Cc) / 256, 256, 0, stream>>>(xn, zp);
  k_gate<<<Bb, 256, 0, stream>>>(zp, fc1_w, fc1_b, fc2_w, fc2_b, gate);
  // 3) in-proj GEMM (WMMA) + split/SiLU
  k_gemm_inproj<<<dim3(Ll / 128, 128 / 16, Gg * Bb), 256, 0, stream>>>(xn, in_w, xc, zb);
  // 4) depthwise conv + SiLU -> u
  k_conv_dw<<<(Gg * Bb * Ll * Dd) / 256, 256, 0, stream>>>(xc, conv_w, conv_b, ub);
  // 5) x_dbl projections (dt_raw, Bs, Cs)
  k_xdbl<<<(Gg * TOK) / 8, 256, 0, stream>>>(ub, xproj_w, dbc);
  // 6-8) chunked selective scan
  k_scan_a<<<Gg * Bb * NCH, CHL, 0, stream>>>(ub, dbc, dtproj_w, dtproj_b, A_log, chP, chS);
  k_scan_b<<<(Gg * Bb * Dd) / 256, 256, 0, stream>>>(chP, chS, h0b);
  k_scan_c<<<Gg * Bb * NCH, CHL, 0, stream>>>(ub, dbc, dtproj_w, dtproj_b, A_log,
                                              Dskip, h0b, yb);
  // 9) group LayerNorm * silu(z)
  k_ln_group<<<(Gg * TOK) / 8, 256, 0, stream>>>(yb, zb, onorm_w, onorm_b);
  // 10) out-proj GEMM (WMMA)
  k_gemm_outproj<<<dim3(TOK / 128, Dd / 16, Gg), 256, 0, stream>>>(yb, out_w, ym);
  // 11) skip*gate scaling + final LayerNorm
  k_scale_ln<<<TOK / 8, 256, 0, stream>>>(ym, xn, gate, skip_s, norm_w, norm_b, tn);
  // 12) final projection GEMM (WMMA) + bias
  k_gemm_final<<<dim3(TOK / 128, Cc / 16), 256, 0, stream>>>(tn, proj_w, proj_b, out);
}